// PointGroupV2_45406394253436
// MI455X (gfx1250) — compile-verified
//
#include <hip/hip_runtime.h>

typedef float v2f __attribute__((ext_vector_type(2)));
typedef float v8f __attribute__((ext_vector_type(8)));

#define C64 64
#define BMAX 16
#define TILE_ROWS 64
#define PROJ_THREADS 128
#define ENCNEGINF 0x007FFFFFu   // encf(-inf)

__device__ __forceinline__ unsigned encf(float f) {
    unsigned u = __float_as_uint(f);
    return (u & 0x80000000u) ? ~u : (u | 0x80000000u);
}
__device__ __forceinline__ float decf(unsigned u) {
    unsigned b = (u & 0x80000000u) ? (u ^ 0x80000000u) : ~u;
    return __uint_as_float(b);
}

// Paired W^T LDS layout: element (k,n) of W^T (= W[n][k]) stored at
// [(k>>1)*128 + n*2 + (k&1)] so a lane's B fragment {k,k+1} is one aligned float2.
__device__ __forceinline__ int wt_idx(int k, int n) {
    return ((k >> 1) << 7) + (n << 1) + (k & 1);
}

// ---------------------------------------------------------------------------
// K0: project k,v (B rows, tiny), fold bias + 1/sqrt(d) scale into kps;
//     initialize segment max (encoded) and segment sums.
// ---------------------------------------------------------------------------
__global__ void k_prep(const float* __restrict__ kin, const float* __restrict__ vin,
                       const float* __restrict__ Wk, const float* __restrict__ bk,
                       const float* __restrict__ Wv, const float* __restrict__ bv,
                       float* __restrict__ kps, float* __restrict__ vp,
                       unsigned* __restrict__ msegU, float* __restrict__ zseg,
                       int B) {
    const int t = threadIdx.x;
    if (t < BMAX * C64) { msegU[t] = ENCNEGINF; zseg[t] = 0.0f; }
    if (t < B * C64) {
        const int b = t >> 6, c = t & 63;
        const float scale = 0.35355339059327373f;   // 1/sqrt(8)
        float accK = bk[c], accV = bv[c];
        for (int j = 0; j < C64; ++j) {
            accK += kin[b * C64 + j] * Wk[c * C64 + j];
            accV += vin[b * C64 + j] * Wv[c * C64 + j];
        }
        kps[t] = accK * scale;
        vp[t]  = accV;
    }
}

// ---------------------------------------------------------------------------
// K1: qp = q @ Wq^T + bq via V_WMMA_F32_16X16X4_F32; attn = qp * kps[seg];
//     store attn; segment max via LDS atomics -> sparse global atomicMax.
// ---------------------------------------------------------------------------
__global__ void k_proj_attn(const float* __restrict__ q,
                            const float* __restrict__ Wq,
                            const float* __restrict__ bq,
                            const int* __restrict__ batch,
                            const float* __restrict__ kps,
                            float* __restrict__ attn,
                            unsigned* __restrict__ gmaxU,
                            int N, int B) {
    __shared__ float    sWT[C64 * C64];      // paired layout (wt_idx)
    __shared__ float    sQ[TILE_ROWS * C64];
    __shared__ float    sKps[BMAX * C64];
    __shared__ float    sBq[C64];
    __shared__ unsigned sMax[BMAX * C64];
    __shared__ int      sSeg[TILE_ROWS];

    const int tid  = threadIdx.x;
    const int row0 = blockIdx.x * TILE_ROWS;

    for (int i = tid; i < C64 * C64; i += PROJ_THREADS) {
        int n = i >> 6, k = i & 63;
        sWT[wt_idx(k, n)] = Wq[i];           // coalesced read, paired store
    }
    for (int i = tid; i < TILE_ROWS * 16; i += PROJ_THREADS) {
        int r = i >> 4, c4 = (i & 15) << 2;
        int row = row0 + r;
        float4 val = make_float4(0.f, 0.f, 0.f, 0.f);
        if (row < N) val = *(const float4*)(q + (long long)row * C64 + c4);
        *(float4*)(sQ + r * C64 + c4) = val;
    }
    for (int i = tid; i < B * C64; i += PROJ_THREADS) sKps[i] = kps[i];
    for (int i = tid; i < BMAX * C64; i += PROJ_THREADS) sMax[i] = ENCNEGINF;
    if (tid < C64) sBq[tid] = bq[tid];
    for (int i = tid; i < TILE_ROWS; i += PROJ_THREADS) {
        int row = row0 + i;
        sSeg[i] = (row < N) ? batch[row] : 0;
    }
    __syncthreads();

    const int wave  = tid >> 5;              // wave32: 4 waves, 16 rows each
    const int lane  = tid & 31;
    const int lhalf = lane >> 4;
    const int lmod  = lane & 15;

    v8f acc[4];
    acc[0] = {0.f,0.f,0.f,0.f,0.f,0.f,0.f,0.f};
    acc[1] = acc[0]; acc[2] = acc[0]; acc[3] = acc[0];

    const int mbase = wave * 16 + lmod;      // A: m = lane%16
    for (int k0 = 0; k0 < C64; k0 += 4) {
        const int kk = k0 + (lhalf << 1);    // A/B: k = k0 + 2*(lane/16) + j
        v2f a = *(const v2f*)(sQ + mbase * C64 + kk);           // aligned b64
        const float* brow = sWT + ((kk >> 1) << 7) + (lmod << 1);
#pragma unroll
        for (int t = 0; t < 4; ++t) {
            v2f b = *(const v2f*)(brow + t * 32);                // aligned b64
            acc[t] = __builtin_amdgcn_wmma_f32_16x16x4_f32(
                false, a, false, b, (short)0, acc[t], false, false);
        }
    }

    // Epilogue: D layout m = reg + 8*(lane/16), n = lane%16
#pragma unroll
    for (int t = 0; t < 4; ++t) {
        const int   c    = t * 16 + lmod;
        const float bias = sBq[c];
#pragma unroll
        for (int r = 0; r < 8; ++r) {
            const int m   = wave * 16 + r + (lhalf << 3);
            const int row = row0 + m;
            if (row < N) {
                const int seg = sSeg[m];
                float val = (acc[t][r] + bias) * sKps[seg * C64 + c];
                attn[(long long)row * C64 + c] = val;
                atomicMax(&sMax[seg * C64 + c], encf(val));
            }
        }
    }
    __syncthreads();
    for (int i = tid; i < B * C64; i += PROJ_THREADS) {
        unsigned u = sMax[i];
        if (u != ENCNEGINF) atomicMax(&gmaxU[i], u);
    }
}

// ---------------------------------------------------------------------------
// K1.5: decode max.  K2.5: reciprocal of z.
// ---------------------------------------------------------------------------
__global__ void k_final1(const unsigned* __restrict__ msegU, float* __restrict__ msegF) {
    int t = threadIdx.x;
    if (t < BMAX * C64) msegF[t] = decf(msegU[t]);
}
__global__ void k_final2(const float* __restrict__ zseg, float* __restrict__ rzseg) {
    int t = threadIdx.x;
    if (t < BMAX * C64) {
        float z = zseg[t];
        rzseg[t] = (z != 0.0f) ? (1.0f / z) : 0.0f;
    }
}

// ---------------------------------------------------------------------------
// K2: z[seg][c] = sum exp(attn - m).  Thread-private accumulation (channel
// group is invariant across a thread's iterations; segment nearly so since
// batch is sorted) -> LDS float atomics -> global atomicAdd.
// ---------------------------------------------------------------------------
__global__ void k_sumexp(const float* __restrict__ attn,
                         const int* __restrict__ batch,
                         const float* __restrict__ msegF,
                         float* __restrict__ zseg, int N, int B) {
    __shared__ float sM[BMAX * C64];
    __shared__ float sZ[BMAX * C64];
    const int tid = threadIdx.x;     // 256 threads; 4 float4 each -> 64 rows/block
    for (int i = tid; i < B * C64; i += 256) sM[i] = msegF[i];
    for (int i = tid; i < BMAX * C64; i += 256) sZ[i] = 0.0f;
    __syncthreads();

    const long long base = (long long)blockIdx.x * 1024;   // float4 index
    const int c4 = (int)((base + tid) & 15) << 2;          // invariant per thread
    float a0 = 0.f, a1 = 0.f, a2 = 0.f, a3 = 0.f;
    int curseg = -1;
#pragma unroll
    for (int it = 0; it < 4; ++it) {
        const long long idx = base + it * 256 + tid;       // idx&15 constant
        const long long row = idx >> 4;
        if (row < N) {
            const int seg = batch[row];
            if (seg != curseg) {
                if (curseg >= 0) {
                    const int o = curseg * C64 + c4;
                    atomicAdd(&sZ[o + 0], a0); atomicAdd(&sZ[o + 1], a1);
                    atomicAdd(&sZ[o + 2], a2); atomicAdd(&sZ[o + 3], a3);
                }
                a0 = a1 = a2 = a3 = 0.f;
                curseg = seg;
            }
            const int o = seg * C64 + c4;
            float4 av = *(const float4*)(attn + row * C64 + c4);
            a0 += __expf(av.x - sM[o + 0]);
            a1 += __expf(av.y - sM[o + 1]);
            a2 += __expf(av.z - sM[o + 2]);
            a3 += __expf(av.w - sM[o + 3]);
        }
    }
    if (curseg >= 0) {
        const int o = curseg * C64 + c4;
        atomicAdd(&sZ[o + 0], a0); atomicAdd(&sZ[o + 1], a1);
        atomicAdd(&sZ[o + 2], a2); atomicAdd(&sZ[o + 3], a3);
    }
    __syncthreads();
    for (int i = tid; i < B * C64; i += 256) {
        float s = sZ[i];
        if (s != 0.0f) atomicAdd(&zseg[i], s);
    }
}

// ---------------------------------------------------------------------------
// K3: w = exp(attn-m)*rz*vp[seg]; out = w @ Wo^T + bo via WMMA.
// ---------------------------------------------------------------------------
__global__ void k_out(const float* __restrict__ attn,
                      const int* __restrict__ batch,
                      const float* __restrict__ Wo,
                      const float* __restrict__ bo,
                      const float* __restrict__ vp,
                      const float* __restrict__ msegF,
                      const float* __restrict__ rzseg,
                      float* __restrict__ out, int N, int B) {
    __shared__ float sWT[C64 * C64];     // paired layout (wt_idx)
    __shared__ float sW[TILE_ROWS * C64];
    __shared__ float sVp[BMAX * C64];
    __shared__ float sM[BMAX * C64];
    __shared__ float sRz[BMAX * C64];
    __shared__ float sBo[C64];

    const int tid  = threadIdx.x;
    const int row0 = blockIdx.x * TILE_ROWS;

    for (int i = tid; i < C64 * C64; i += PROJ_THREADS) {
        int n = i >> 6, k = i & 63;
        sWT[wt_idx(k, n)] = Wo[i];
    }
    for (int i = tid; i < B * C64; i += PROJ_THREADS) {
        sVp[i] = vp[i]; sM[i] = msegF[i]; sRz[i] = rzseg[i];
    }
    if (tid < C64) sBo[tid] = bo[tid];
    __syncthreads();

    // build weighted-value tile in LDS
    for (int i = tid; i < TILE_ROWS * 16; i += PROJ_THREADS) {
        const int r = i >> 4, c4 = (i & 15) << 2;
        const int row = row0 + r;
        float4 res = make_float4(0.f, 0.f, 0.f, 0.f);
        if (row < N) {
            const int seg = batch[row];
            float4 av = *(const float4*)(attn + (long long)row * C64 + c4);
            const int o = seg * C64 + c4;
            res.x = __expf(av.x - sM[o + 0]) * sRz[o + 0] * sVp[o + 0];
            res.y = __expf(av.y - sM[o + 1]) * sRz[o + 1] * sVp[o + 1];
            res.z = __expf(av.z - sM[o + 2]) * sRz[o + 2] * sVp[o + 2];
            res.w = __expf(av.w - sM[o + 3]) * sRz[o + 3] * sVp[o + 3];
        }
        *(float4*)(sW + r * C64 + c4) = res;
    }
    __syncthreads();

    const int wave  = tid >> 5;
    const int lane  = tid & 31;
    const int lhalf = lane >> 4;
    const int lmod  = lane & 15;

    v8f acc[4];
    acc[0] = {0.f,0.f,0.f,0.f,0.f,0.f,0.f,0.f};
    acc[1] = acc[0]; acc[2] = acc[0]; acc[3] = acc[0];

    const int mbase = wave * 16 + lmod;
    for (int k0 = 0; k0 < C64; k0 += 4) {
        const int kk = k0 + (lhalf << 1);
        v2f a = *(const v2f*)(sW + mbase * C64 + kk);
        const float* brow = sWT + ((kk >> 1) << 7) + (lmod << 1);
#pragma unroll
        for (int t = 0; t < 4; ++t) {
            v2f b = *(const v2f*)(brow + t * 32);
            acc[t] = __builtin_amdgcn_wmma_f32_16x16x4_f32(
                false, a, false, b, (short)0, acc[t], false, false);
        }
    }

#pragma unroll
    for (int t = 0; t < 4; ++t) {
        const int   c    = t * 16 + lmod;
        const float bias = sBo[c];
#pragma unroll
        for (int r = 0; r < 8; ++r) {
            const int m   = wave * 16 + r + (lhalf << 3);
            const int row = row0 + m;
            if (row < N) out[(long long)row * C64 + c] = acc[t][r] + bias;
        }
    }
}

// ---------------------------------------------------------------------------
extern "C" void kernel_launch(void* const* d_in, const int* in_sizes, int n_in,
                              void* d_out, int out_size, void* d_ws, size_t ws_size,
                              hipStream_t stream) {
    const float* q     = (const float*)d_in[0];
    const float* kin   = (const float*)d_in[1];
    const float* vin   = (const float*)d_in[2];
    const float* Wq    = (const float*)d_in[3];
    const float* bq    = (const float*)d_in[4];
    const float* Wk    = (const float*)d_in[5];
    const float* bk    = (const float*)d_in[6];
    const float* Wv    = (const float*)d_in[7];
    const float* bv    = (const float*)d_in[8];
    const float* Wo    = (const float*)d_in[9];
    const float* bo    = (const float*)d_in[10];
    const int*   batch = (const int*)d_in[11];
    float*       out   = (float*)d_out;

    const int N = in_sizes[0] / C64;
    const int B = in_sizes[1] / C64;

    char* ws = (char*)d_ws;
    float*    kps   = (float*)(ws);
    float*    vp    = (float*)(ws + 4096);
    unsigned* msegU = (unsigned*)(ws + 8192);
    float*    msegF = (float*)(ws + 12288);
    float*    zseg  = (float*)(ws + 16384);
    float*    rzseg = (float*)(ws + 20480);
    float*    attn  = (float*)(ws + 24576);

    const int nb = (N + TILE_ROWS - 1) / TILE_ROWS;

    k_prep<<<1, 1024, 0, stream>>>(kin, vin, Wk, bk, Wv, bv, kps, vp, msegU, zseg, B);
    k_proj_attn<<<nb, PROJ_THREADS, 0, stream>>>(q, Wq, bq, batch, kps, attn, msegU, N, B);
    k_final1<<<1, 1024, 0, stream>>>(msegU, msegF);
    k_sumexp<<<nb, 256, 0, stream>>>(attn, batch, msegF, zseg, N, B);
    k_final2<<<1, 1024, 0, stream>>>(zseg, rzseg);
    k_out<<<nb, PROJ_THREADS, 0, stream>>>(attn, batch, Wo, bo, vp, msegF, rzseg, out, N, B);
}